// GRUDCell_73864847557195
// MI455X (gfx1250) — compile-verified
//
#include <hip/hip_runtime.h>
#include <math.h>

typedef __attribute__((ext_vector_type(2))) float v2f;
typedef __attribute__((ext_vector_type(4))) float v4f;
typedef __attribute__((ext_vector_type(8))) float v8f;

#define B_   8192
#define I_   512
#define H_   1024
#define D_   8
#define KTOT (I_ + H_)          // 1536 virtual K = [x | gamma*h]
#define BM   64
#define BN   128
#define BK   16
#define LSTR 20                 // LDS row stride (floats): 80B -> 16B aligned, bank-conflict free
#define NCHUNK (KTOT / BK)      // 96
#define NCX   (I_ / BK)         // 32 chunks in the x region

// float offsets inside one LDS stage
#define OFF_A   0
#define OFF_W0  (BM * LSTR)                 // 1280
#define OFF_W1  (OFF_W0 + BN * LSTR)        // 3840
#define OFF_W2  (OFF_W1 + BN * LSTR)        // 6400
#define STAGE   (OFF_W2 + BN * LSTR)        // 8960 floats = 35840 B per stage

// async copy of 16 bytes global -> LDS, tracked by ASYNCcnt (no VGPR data path)
__device__ __forceinline__ void async_b128(unsigned lds_byte_addr, const void* gptr) {
    asm volatile("global_load_async_to_lds_b128 %0, %1, off"
                 :: "v"(lds_byte_addr), "v"((unsigned long long)gptr)
                 : "memory");
}
__device__ __forceinline__ void wait_async0() {
    asm volatile("s_wait_asynccnt 0" ::: "memory");
}

// gamma_h(b, j) = exp(-relu(delta[b,:] . w_gamma[j,:] + b_gamma[j]))
__device__ __forceinline__ float gamma_decay(const float* __restrict__ delta,
                                             const float* __restrict__ w_gamma,
                                             const float* __restrict__ b_gamma,
                                             int b, int j) {
    const v4f* dv = (const v4f*)(delta + (size_t)b * D_);
    const v4f* wv = (const v4f*)(w_gamma + (size_t)j * D_);
    v4f d0 = dv[0], d1 = dv[1], w0 = wv[0], w1 = wv[1];
    float s = d0.x * w0.x + d0.y * w0.y + d0.z * w0.z + d0.w * w0.w
            + d1.x * w1.x + d1.y * w1.y + d1.z * w1.z + d1.w * w1.w
            + b_gamma[j];
    s = s > 0.0f ? s : 0.0f;
    return __expf(-s);
}

__device__ __forceinline__ float sigmoidf_(float v) {
    return 1.0f / (1.0f + __expf(-v));
}

// one BK=16 chunk of WMMA work on stage `sb`; branch-free, accn selects nx/nh set
__device__ __forceinline__ void wmma_chunk(const float* sb, int wr, int wc, int lr, int hi,
                                           v8f (&acc_r)[2][2], v8f (&acc_z)[2][2],
                                           v8f (&accn)[2][2]) {
    const float* sA  = sb + OFF_A;
    const float* sW0 = sb + OFF_W0;
    const float* sW1 = sb + OFF_W1;
    const float* sW2 = sb + OFF_W2;
#pragma unroll
    for (int kk = 0; kk < BK; kk += 4) {
        const int ko = kk + 2 * hi;   // lane-half k split per ISA A/B layout
        v2f a0 = *(const v2f*)(sA + (wr * 32 +      lr) * LSTR + ko);
        v2f a1 = *(const v2f*)(sA + (wr * 32 + 16 + lr) * LSTR + ko);

        v2f br0 = *(const v2f*)(sW0 + (wc * 32 +      lr) * LSTR + ko);
        v2f br1 = *(const v2f*)(sW0 + (wc * 32 + 16 + lr) * LSTR + ko);
        acc_r[0][0] = __builtin_amdgcn_wmma_f32_16x16x4_f32(false, a0, false, br0, (short)0, acc_r[0][0], false, false);
        acc_r[0][1] = __builtin_amdgcn_wmma_f32_16x16x4_f32(false, a0, false, br1, (short)0, acc_r[0][1], false, false);
        acc_r[1][0] = __builtin_amdgcn_wmma_f32_16x16x4_f32(false, a1, false, br0, (short)0, acc_r[1][0], false, false);
        acc_r[1][1] = __builtin_amdgcn_wmma_f32_16x16x4_f32(false, a1, false, br1, (short)0, acc_r[1][1], false, false);

        v2f bz0 = *(const v2f*)(sW1 + (wc * 32 +      lr) * LSTR + ko);
        v2f bz1 = *(const v2f*)(sW1 + (wc * 32 + 16 + lr) * LSTR + ko);
        acc_z[0][0] = __builtin_amdgcn_wmma_f32_16x16x4_f32(false, a0, false, bz0, (short)0, acc_z[0][0], false, false);
        acc_z[0][1] = __builtin_amdgcn_wmma_f32_16x16x4_f32(false, a0, false, bz1, (short)0, acc_z[0][1], false, false);
        acc_z[1][0] = __builtin_amdgcn_wmma_f32_16x16x4_f32(false, a1, false, bz0, (short)0, acc_z[1][0], false, false);
        acc_z[1][1] = __builtin_amdgcn_wmma_f32_16x16x4_f32(false, a1, false, bz1, (short)0, acc_z[1][1], false, false);

        v2f bn0 = *(const v2f*)(sW2 + (wc * 32 +      lr) * LSTR + ko);
        v2f bn1 = *(const v2f*)(sW2 + (wc * 32 + 16 + lr) * LSTR + ko);
        accn[0][0] = __builtin_amdgcn_wmma_f32_16x16x4_f32(false, a0, false, bn0, (short)0, accn[0][0], false, false);
        accn[0][1] = __builtin_amdgcn_wmma_f32_16x16x4_f32(false, a0, false, bn1, (short)0, accn[0][1], false, false);
        accn[1][0] = __builtin_amdgcn_wmma_f32_16x16x4_f32(false, a1, false, bn0, (short)0, accn[1][0], false, false);
        accn[1][1] = __builtin_amdgcn_wmma_f32_16x16x4_f32(false, a1, false, bn1, (short)0, accn[1][1], false, false);
    }
}

__global__ __launch_bounds__(256)
void grud_fused_wmma(const float* __restrict__ x,
                     const float* __restrict__ delta,
                     const float* __restrict__ h,
                     const float* __restrict__ w_ih,   // [3H, I]
                     const float* __restrict__ w_hh,   // [3H, H]
                     const float* __restrict__ b_ih,   // [3H]
                     const float* __restrict__ b_hh,   // [3H]
                     const float* __restrict__ w_gamma,// [H, D]
                     const float* __restrict__ b_gamma,// [H]
                     float* __restrict__ out)          // [B, H]
{
    __shared__ float smem[2 * STAGE];   // double-buffered stages

    const int tid  = threadIdx.x;
    const int lane = tid & 31;
    const int wv   = tid >> 5;     // wave 0..7
    const int wr   = wv & 1;       // wave row (0..1)  -> 32 rows each
    const int wc   = wv >> 1;      // wave col (0..3)  -> 32 cols each
    const int lr   = lane & 15;
    const int hi   = lane >> 4;

    const int rowBase = blockIdx.y * BM;   // batch rows
    const int colBase = blockIdx.x * BN;   // hidden cols

    // low 32 bits of the generic pointer == LDS byte offset (aperture rule)
    const unsigned ldsBase = (unsigned)(size_t)(void*)smem;

    v8f acc_r[2][2], acc_z[2][2], acc_nx[2][2], acc_nh[2][2];
#pragma unroll
    for (int t = 0; t < 2; ++t)
#pragma unroll
        for (int u = 0; u < 2; ++u) {
            acc_r[t][u]  = (v8f)0.0f;
            acc_z[t][u]  = (v8f)0.0f;
            acc_nx[t][u] = (v8f)0.0f;
            acc_nh[t][u] = (v8f)0.0f;
        }

    // per-thread staging coordinates: 1 float4 of A, 2 float4 per gate of W
    const int aRow = tid >> 2;            // 0..63
    const int f4   = (tid & 3) << 2;      // k offset within chunk: 0,4,8,12
    const unsigned wOff[3] = {OFF_W0, OFF_W1, OFF_W2};

    // ---- prologue: async-fill stage 0 with chunk 0 (x region) ----
    async_b128(ldsBase + (OFF_A + aRow * LSTR + f4) * 4u,
               x + (size_t)(rowBase + aRow) * I_ + f4);
#pragma unroll
    for (int g = 0; g < 3; ++g)
#pragma unroll
        for (int rep = 0; rep < 2; ++rep) {
            const int grow = g * H_ + colBase + aRow + rep * 64;
            async_b128(ldsBase + (wOff[g] + (aRow + rep * 64) * LSTR + f4) * 4u,
                       w_ih + (size_t)grow * I_ + f4);
        }
    wait_async0();
    __syncthreads();

    int p = 0;

    // ================= phase X: chunks 0..31, n-gate -> acc_nx =================
    for (int c = 0; c < NCX; ++c) {
        const int  k1     = (c + 1) * BK;       // chunk to prefetch
        const bool pfX    = (k1 < I_);          // false only when c == 31
        const unsigned nb = (unsigned)(p ^ 1) * STAGE;

        // ---- prefetch chunk c+1 into the other stage ----
#pragma unroll
        for (int g = 0; g < 3; ++g)
#pragma unroll
            for (int rep = 0; rep < 2; ++rep) {
                const int grow = g * H_ + colBase + aRow + rep * 64;
                const void* gp = pfX
                    ? (const void*)(w_ih + (size_t)grow * I_ + k1 + f4)
                    : (const void*)(w_hh + (size_t)grow * H_ + (k1 - I_) + f4);
                async_b128(ldsBase + (nb + wOff[g] + (aRow + rep * 64) * LSTR + f4) * 4u, gp);
            }
        v4f  aval;
        bool needAStore = false;
        if (pfX) {
            async_b128(ldsBase + (nb + OFF_A + aRow * LSTR + f4) * 4u,
                       x + (size_t)(rowBase + aRow) * I_ + k1 + f4);
        } else {  // first h-region chunk: fuse gamma decay on the VGPR path
            const int bb = rowBase + aRow;
            const int kc = k1 - I_ + f4;
            aval = *(const v4f*)(h + (size_t)bb * H_ + kc);
            aval.x *= gamma_decay(delta, w_gamma, b_gamma, bb, kc + 0);
            aval.y *= gamma_decay(delta, w_gamma, b_gamma, bb, kc + 1);
            aval.z *= gamma_decay(delta, w_gamma, b_gamma, bb, kc + 2);
            aval.w *= gamma_decay(delta, w_gamma, b_gamma, bb, kc + 3);
            needAStore = true;
        }

        wmma_chunk(smem + (unsigned)p * STAGE, wr, wc, lr, hi, acc_r, acc_z, acc_nx);

        if (needAStore)   // uniform
            *(v4f*)(smem + nb + OFF_A + aRow * LSTR + f4) = aval;
        wait_async0();
        __syncthreads();
        p ^= 1;
    }

    // ================= phase H: chunks 32..95, n-gate -> acc_nh =================
    for (int c = NCX; c < NCHUNK; ++c) {
        const bool havePf = (c + 1 < NCHUNK);
        const int  k1     = (c + 1) * BK;
        const unsigned nb = (unsigned)(p ^ 1) * STAGE;

        v4f  aval;
        if (havePf) {
#pragma unroll
            for (int g = 0; g < 3; ++g)
#pragma unroll
                for (int rep = 0; rep < 2; ++rep) {
                    const int grow = g * H_ + colBase + aRow + rep * 64;
                    async_b128(ldsBase + (nb + wOff[g] + (aRow + rep * 64) * LSTR + f4) * 4u,
                               w_hh + (size_t)grow * H_ + (k1 - I_) + f4);
                }
            const int bb = rowBase + aRow;
            const int kc = k1 - I_ + f4;
            aval = *(const v4f*)(h + (size_t)bb * H_ + kc);
            aval.x *= gamma_decay(delta, w_gamma, b_gamma, bb, kc + 0);
            aval.y *= gamma_decay(delta, w_gamma, b_gamma, bb, kc + 1);
            aval.z *= gamma_decay(delta, w_gamma, b_gamma, bb, kc + 2);
            aval.w *= gamma_decay(delta, w_gamma, b_gamma, bb, kc + 3);
        }

        wmma_chunk(smem + (unsigned)p * STAGE, wr, wc, lr, hi, acc_r, acc_z, acc_nh);

        if (havePf)   // uniform
            *(v4f*)(smem + nb + OFF_A + aRow * LSTR + f4) = aval;
        wait_async0();
        __syncthreads();
        p ^= 1;
    }

    // ---- fused gate epilogue ----
#pragma unroll
    for (int u = 0; u < 2; ++u) {
        const int j   = colBase + wc * 32 + u * 16 + lr;
        const float brz = b_ih[j] + b_hh[j];
        const float bzz = b_ih[H_ + j] + b_hh[H_ + j];
        const float bin = b_ih[2 * H_ + j];
        const float bhn = b_hh[2 * H_ + j];
#pragma unroll
        for (int t = 0; t < 2; ++t) {
#pragma unroll
            for (int v = 0; v < 8; ++v) {
                const int i = rowBase + wr * 32 + t * 16 + v + 8 * hi;
                const float hp = gamma_decay(delta, w_gamma, b_gamma, i, j)
                               * h[(size_t)i * H_ + j];
                const float r = sigmoidf_(acc_r[t][u][v] + brz);
                const float z = sigmoidf_(acc_z[t][u][v] + bzz);
                const float n = tanhf(acc_nx[t][u][v] + bin
                                      + r * (acc_nh[t][u][v] + bhn));
                out[(size_t)i * H_ + j] = n + z * (hp - n);
            }
        }
    }
}

extern "C" void kernel_launch(void* const* d_in, const int* in_sizes, int n_in,
                              void* d_out, int out_size, void* d_ws, size_t ws_size,
                              hipStream_t stream) {
    (void)in_sizes; (void)n_in; (void)out_size; (void)d_ws; (void)ws_size;
    const float* x       = (const float*)d_in[0];
    const float* delta   = (const float*)d_in[1];
    const float* h       = (const float*)d_in[2];
    const float* w_ih    = (const float*)d_in[3];
    const float* w_hh    = (const float*)d_in[4];
    const float* b_ih    = (const float*)d_in[5];
    const float* b_hh    = (const float*)d_in[6];
    const float* w_gamma = (const float*)d_in[7];
    const float* b_gamma = (const float*)d_in[8];
    float* out = (float*)d_out;

    dim3 grid(H_ / BN, B_ / BM);   // (8, 128)
    dim3 block(256);
    grud_fused_wmma<<<grid, block, 0, stream>>>(x, delta, h, w_ih, w_hh,
                                                b_ih, b_hh, w_gamma, b_gamma, out);
}